// StaticGCN_15659450761585
// MI455X (gfx1250) — compile-verified
//
#include <hip/hip_runtime.h>

// ---------------------------------------------------------------------------
// StaticGCN collapsed-form solver for MI455X (gfx1250, wave32).
//
// Math (exact reformulation of the reference, which overwrites x with ones):
//   deg[v]  = 1 + sum_{e: col=v} w[e]
//   dinv[v] = deg>0 ? rsqrt(deg) : 0
//   c[v]    = dinv[v]^2 + sum_{e: col=v} dinv[row]*w*dinv[v]     (incoming norm)
//   r[v]    = dinv[v]^2 + sum_{e: row=v} dinv[v]*w*dinv[col]     (outgoing norm)
//   s1[j]   = sum_i W1[i][j]
//   u[j]    = sum_v r[v] * relu(c[v]*s1[j] + b1[j])
//   out     = (u @ W2)/N + b2            <- done with V_WMMA_F32_16X16X4_F32
// ---------------------------------------------------------------------------

typedef __attribute__((ext_vector_type(2))) float v2f;
typedef __attribute__((ext_vector_type(8))) float v8f;

#define F 128  // IN == HID == OUT == 128

// --- 1. init: deg = 1 (self-loop weight), zero the u accumulator -----------
__global__ void k_init(float* __restrict__ deg, float* __restrict__ u, int n) {
    int i = blockIdx.x * blockDim.x + threadIdx.x;
    if (i < n)  deg[i] = 1.0f;
    if (i < F)  u[i]   = 0.0f;
}

// --- 2. degree scatter: deg[col[e]] += w[e] --------------------------------
__global__ void k_deg(const int* __restrict__ col, const float* __restrict__ w,
                      float* __restrict__ deg, int e) {
    int i = blockIdx.x * blockDim.x + threadIdx.x;
    if (i < e) atomicAdd(&deg[col[i]], w[i]);
}

// --- 3. dinv + seed c,r with self-loop norm dinv^2 -------------------------
__global__ void k_dinv(const float* __restrict__ deg, float* __restrict__ dinv,
                       float* __restrict__ c, float* __restrict__ r, int n) {
    int i = blockIdx.x * blockDim.x + threadIdx.x;
    if (i < n) {
        float d  = deg[i];
        float di = (d > 0.0f) ? rsqrtf(d) : 0.0f;
        dinv[i]  = di;
        float d2 = di * di;
        c[i] = d2;
        r[i] = d2;
    }
}

// --- 4. column sums of W1: s1[j] = sum_i W1[i*F + j] -----------------------
__global__ void k_colsum(const float* __restrict__ W1, float* __restrict__ s1) {
    int j = threadIdx.x;  // one block of F threads, coalesced across j
    float s = 0.0f;
    for (int i = 0; i < F; ++i) s += W1[i * F + j];
    s1[j] = s;
}

// --- 5. edge scatter of norms into c (at dst) and r (at src) ---------------
__global__ void k_edge(const int* __restrict__ row, const int* __restrict__ col,
                       const float* __restrict__ w, const float* __restrict__ dinv,
                       float* __restrict__ c, float* __restrict__ r, int e) {
    int i = blockIdx.x * blockDim.x + threadIdx.x;
    if (i < e) {
        int a = row[i], b = col[i];
        float nrm = dinv[a] * w[i] * dinv[b];
        atomicAdd(&c[b], nrm);   // incoming-norm sum at destination
        atomicAdd(&r[a], nrm);   // outgoing-norm sum at source
    }
}

// --- 6. node accumulation: u[j] += sum_v r[v]*relu(c[v]*s1[j]+b1[j]) -------
__global__ void k_accum(const float* __restrict__ rv, const float* __restrict__ cv,
                        const float* __restrict__ s1, const float* __restrict__ b1,
                        float* __restrict__ u, int n) {
    __shared__ float sm[256];
    int j   = threadIdx.x & (F - 1);  // 0..127 : feature column
    int sub = threadIdx.x >> 7;       // 0..1   : node sub-row within block
    float s  = s1[j];
    float bb = b1[j];
    float acc = 0.0f;
    for (int v = blockIdx.x * 2 + sub; v < n; v += gridDim.x * 2) {
        float g = rv[v];
        float h = fmaf(cv[v], s, bb);
        acc += g * (h > 0.0f ? h : 0.0f);
    }
    sm[threadIdx.x] = acc;
    __syncthreads();
    if (sub == 0) atomicAdd(&u[j], sm[j] + sm[j + F]);
}

// --- 7. final matvec via V_WMMA_F32_16X16X4_F32: out = (u @ W2)/N + b2 -----
// One 256-thread block = 8 waves; wave w owns output columns [16w, 16w+16).
// A fragment (16x4, ISA 7.12.2): lane holds row M=lane%16; VGPR0 <-> K=2*(lane>=16),
// VGPR1 <-> K+1. We replicate u across all M rows so D row 0 carries the result.
// B fragment (4x16) mirrors A with N = lane%16.
__global__ void k_final(const float* __restrict__ u, const float* __restrict__ W2,
                        const float* __restrict__ b2, float* __restrict__ out,
                        float invN) {
    int lane = threadIdx.x & 31;
    int wv   = threadIdx.x >> 5;   // wave id 0..7 -> n-block
    int half = lane >> 4;          // 0: lanes 0-15, 1: lanes 16-31
    int l15  = lane & 15;
    v8f acc = {};
    #pragma unroll
    for (int kb = 0; kb < 32; ++kb) {
        int k0 = kb * 4 + half * 2;          // K index held in VGPR0 of this lane
        v2f a, b;
        a.x = u[k0];
        a.y = u[k0 + 1];
        b.x = W2[(k0)     * F + wv * 16 + l15];
        b.y = W2[(k0 + 1) * F + wv * 16 + l15];
        acc = __builtin_amdgcn_wmma_f32_16x16x4_f32(
            /*neg_a=*/false, a, /*neg_b=*/false, b,
            /*c_mod=*/(short)0, acc, /*reuse_a=*/false, /*reuse_b=*/false);
    }
    // D layout: VGPR0, lanes 0-15 = row M=0, N=lane  (u replicated over M)
    if (lane < 16) {
        int j = wv * 16 + l15;
        out[j] = acc[0] * invN + b2[j];
    }
}

// ---------------------------------------------------------------------------
extern "C" void kernel_launch(void* const* d_in, const int* in_sizes, int n_in,
                              void* d_out, int out_size, void* d_ws, size_t ws_size,
                              hipStream_t stream) {
    // Inputs (setup_inputs order): x, edge_index, edge_attr, W1, b1, W2, b2
    const int*   ei  = (const int*)  d_in[1];   // [2, E] int32 (JAX x64 off)
    const float* w   = (const float*)d_in[2];   // [E]
    const float* W1  = (const float*)d_in[3];   // [128,128]
    const float* b1  = (const float*)d_in[4];   // [128]
    const float* W2  = (const float*)d_in[5];   // [128,128]
    const float* b2  = (const float*)d_in[6];   // [128]
    float*       out = (float*)d_out;           // [1,128]

    const int n = in_sizes[0] / F;              // 100000
    const int e = in_sizes[2];                  // 1600000
    const int* row = ei;
    const int* col = ei + e;

    // Workspace layout (floats): deg | dinv | c | r | u(128) | s1(128)
    float* ws   = (float*)d_ws;
    float* deg  = ws;
    float* dinv = ws + (size_t)n;
    float* c    = ws + (size_t)2 * n;
    float* r    = ws + (size_t)3 * n;
    float* u    = ws + (size_t)4 * n;
    float* s1   = u + F;

    const int B = 256;
    k_init  <<<(n + B - 1) / B, B, 0, stream>>>(deg, u, n);
    k_deg   <<<(e + B - 1) / B, B, 0, stream>>>(col, w, deg, e);
    k_dinv  <<<(n + B - 1) / B, B, 0, stream>>>(deg, dinv, c, r, n);
    k_colsum<<<1, F, 0, stream>>>(W1, s1);
    k_edge  <<<(e + B - 1) / B, B, 0, stream>>>(row, col, w, dinv, c, r, e);
    k_accum <<<256, B, 0, stream>>>(r, c, s1, b1, u, n);
    k_final <<<1, B, 0, stream>>>(u, W2, b2, out, 1.0f / (float)n);
    (void)n_in; (void)out_size; (void)ws_size;
}